// Model_27676769256199
// MI455X (gfx1250) — compile-verified
//
#include <hip/hip_runtime.h>
#include <math.h>

// ---------------------------------------------------------------------------
// Sizes (fixed by the reference)
// ---------------------------------------------------------------------------
#define NW 40000
#define NDOC 15000
#define ZW 512           // HEADS * fout = 4*128, both layers
#define FOUT 128
#define HEADS 4

typedef float v2f __attribute__((ext_vector_type(2)));
typedef float v8f __attribute__((ext_vector_type(8)));

// ---------------------------------------------------------------------------
// WMMA GEMM:  Z[M,512] = A[M,K] @ W[K,512]   (f32, V_WMMA_F32_16X16X4_F32)
// block = 256 threads (8 waves). Wave w computes cols [w*64, w*64+64) of the
// 16-row strip at blockIdx.x*16. Each wave holds 4 accumulators (16x16 tiles)
// and reuses one A fragment per k-step across the 4 WMMAs.
// ---------------------------------------------------------------------------
__global__ void gemm_wmma_k(const float* __restrict__ A,
                            const float* __restrict__ W,
                            float* __restrict__ Z, int M, int K) {
    const int lane = threadIdx.x & 31;
    const int wave = threadIdx.x >> 5;
    const int r    = lane & 15;
    const int hi   = lane >> 4;            // 0 or 1
    const int row0 = blockIdx.x * 16;
    const int c0   = wave * 64;

    int arow = row0 + r;
    if (arow > M - 1) arow = M - 1;        // clamp (stores are guarded)
    const float* ap = A + (size_t)arow * K;

    v8f acc0 = {}, acc1 = {}, acc2 = {}, acc3 = {};

    for (int k = 0; k < K; k += 4) {
        v2f a;
        a.x = ap[k + 2 * hi];
        a.y = ap[k + 2 * hi + 1];
        const float* w0 = W + (size_t)(k + 2 * hi) * ZW + c0 + r;
        const float* w1 = w0 + ZW;
        v2f b0v; b0v.x = w0[0];  b0v.y = w1[0];
        v2f b1v; b1v.x = w0[16]; b1v.y = w1[16];
        v2f b2v; b2v.x = w0[32]; b2v.y = w1[32];
        v2f b3v; b3v.x = w0[48]; b3v.y = w1[48];
        acc0 = __builtin_amdgcn_wmma_f32_16x16x4_f32(false, a, false, b0v,
                                                     (short)0, acc0, false, false);
        acc1 = __builtin_amdgcn_wmma_f32_16x16x4_f32(false, a, false, b1v,
                                                     (short)0, acc1, false, false);
        acc2 = __builtin_amdgcn_wmma_f32_16x16x4_f32(false, a, false, b2v,
                                                     (short)0, acc2, false, false);
        acc3 = __builtin_amdgcn_wmma_f32_16x16x4_f32(false, a, false, b3v,
                                                     (short)0, acc3, false, false);
    }

#pragma unroll
    for (int j = 0; j < 8; j++) {
        int row = row0 + hi * 8 + j;
        if (row < M) {
            float* zp = Z + (size_t)row * ZW + c0 + r;
            zp[0]  = acc0[j];
            zp[16] = acc1[j];
            zp[32] = acc2[j];
            zp[48] = acc3[j];
        }
    }
}

// ---------------------------------------------------------------------------
// el/er:  out[n,h] = dot(z[n, h*128 : h*128+128], avec[h, :])
// grid = N blocks, block = 128 (4 waves; wave h handles head h)
// ---------------------------------------------------------------------------
__global__ void attn_coef_k(const float* __restrict__ z,
                            const float* __restrict__ avec,
                            float* __restrict__ out, int N) {
    int n    = blockIdx.x;
    int h    = threadIdx.x >> 5;
    int lane = threadIdx.x & 31;
    const float* zp = z + (size_t)n * ZW + h * FOUT;
    const float* ap = avec + h * FOUT;
    float s = 0.f;
    for (int i = lane; i < FOUT; i += 32) s += zp[i] * ap[i];
    for (int off = 16; off; off >>= 1) s += __shfl_xor(s, off, 32);
    if (lane == 0) out[n * HEADS + h] = s;
}

// ---------------------------------------------------------------------------
// float atomic max via int punning (correct for mixed signs; init = -inf)
// ---------------------------------------------------------------------------
__device__ __forceinline__ void atomicMaxF(float* addr, float val) {
    if (val >= 0.f) atomicMax((int*)addr, __float_as_int(val));
    else            atomicMin((unsigned int*)addr, __float_as_uint(val));
}

__device__ __forceinline__ float lrelu(float x) {
    return x > 0.f ? x : 0.2f * x;
}

__global__ void fill_k(float* p, float v, int n) {
    int i = blockIdx.x * blockDim.x + threadIdx.x;
    if (i < n) p[i] = v;
}

// Pass 1: segment max of leaky-relu logits per (dst, head)
__global__ void edge_max_k(const int* __restrict__ src, const int* __restrict__ dst,
                           int E, const float* __restrict__ el,
                           const float* __restrict__ er, float* __restrict__ m) {
    int e = blockIdx.x * blockDim.x + threadIdx.x;
    if (e >= E) return;
    int s = src[e], d = dst[e];
#pragma unroll
    for (int h = 0; h < HEADS; h++) {
        float v = lrelu(el[s * HEADS + h] + er[d * HEADS + h]);
        atomicMaxF(&m[d * HEADS + h], v);
    }
}

// Pass 2: denom = segment sum of exp(e - m)
__global__ void edge_sum_k(const int* __restrict__ src, const int* __restrict__ dst,
                           int E, const float* __restrict__ el,
                           const float* __restrict__ er,
                           const float* __restrict__ m, float* __restrict__ den) {
    int e = blockIdx.x * blockDim.x + threadIdx.x;
    if (e >= E) return;
    int s = src[e], d = dst[e];
#pragma unroll
    for (int h = 0; h < HEADS; h++) {
        float v = lrelu(el[s * HEADS + h] + er[d * HEADS + h]);
        atomicAdd(&den[d * HEADS + h], expf(v - m[d * HEADS + h]));
    }
}

// Pass 3: acc[dst] += alpha * z[src]   (one block per edge, 128 threads = f)
__global__ void edge_agg_k(const int* __restrict__ src, const int* __restrict__ dst,
                           int E, const float* __restrict__ el,
                           const float* __restrict__ er,
                           const float* __restrict__ m, const float* __restrict__ den,
                           const float* __restrict__ z, float* __restrict__ acc) {
    int e = blockIdx.x;
    int s = src[e], d = dst[e];
    __shared__ float salpha[HEADS];
    if (threadIdx.x < HEADS) {
        int h = threadIdx.x;
        float v  = lrelu(el[s * HEADS + h] + er[d * HEADS + h]);
        float ex = expf(v - m[d * HEADS + h]);
        salpha[h] = ex / fmaxf(den[d * HEADS + h], 1e-9f);
    }
    __syncthreads();
    int f = threadIdx.x;  // 0..127
    const float* zs = z + (size_t)s * ZW;
    float* ad = acc + (size_t)d * ZW;
#pragma unroll
    for (int h = 0; h < HEADS; h++)
        atomicAdd(&ad[h * FOUT + f], salpha[h] * zs[h * FOUT + f]);
}

// ---------------------------------------------------------------------------
// finalize: out[n,f] = relu( sum_h acc[n,h,f] + sum_rel sum_h b_rel[h,f] )
// ---------------------------------------------------------------------------
__global__ void finalize_k(const float* __restrict__ acc,
                           const float* __restrict__ b0,
                           const float* __restrict__ b1,
                           const float* __restrict__ b2,
                           float* __restrict__ out, int N) {
    int i = blockIdx.x * blockDim.x + threadIdx.x;
    if (i >= N * FOUT) return;
    int n = i >> 7, f = i & (FOUT - 1);
    float bias = 0.f;
#pragma unroll
    for (int h = 0; h < HEADS; h++) {
        bias += b0[h * FOUT + f];
        if (b1) bias += b1[h * FOUT + f];
        if (b2) bias += b2[h * FOUT + f];
    }
    const float* a = acc + (size_t)n * ZW;
    float s = a[f] + a[FOUT + f] + a[2 * FOUT + f] + a[3 * FOUT + f] + bias;
    out[i] = fmaxf(s, 0.f);
}

// out[d] = sigmoid( dot(hd[d], lin_w) + lin_b ) ; one wave per doc
__global__ void final_linear_k(const float* __restrict__ hd,
                               const float* __restrict__ w,
                               const float* __restrict__ b,
                               float* __restrict__ out, int N) {
    int d = blockIdx.x;
    int lane = threadIdx.x;
    float s = 0.f;
    for (int i = lane; i < FOUT; i += 32) s += hd[(size_t)d * FOUT + i] * w[i];
    for (int off = 16; off; off >>= 1) s += __shfl_xor(s, off, 32);
    if (lane == 0) out[d] = 1.f / (1.f + expf(-(s + b[0])));
}

// ---------------------------------------------------------------------------
// Host-side orchestration
// ---------------------------------------------------------------------------
struct Gat { const float *W, *al, *ar, *b; };

static void run_relation(const float* h_src, const float* h_dst,
                         int Ns, int Nd, int K, const Gat& g,
                         const int* src, const int* dst, int E,
                         float* z_src, float* z_dst,
                         float* el, float* er, float* m, float* den,
                         float* acc, hipStream_t stream) {
    gemm_wmma_k<<<(Ns + 15) / 16, 256, 0, stream>>>(h_src, g.W, z_src, Ns, K);
    if (z_dst != z_src)
        gemm_wmma_k<<<(Nd + 15) / 16, 256, 0, stream>>>(h_dst, g.W, z_dst, Nd, K);
    attn_coef_k<<<Ns, 128, 0, stream>>>(z_src, g.al, el, Ns);
    attn_coef_k<<<Nd, 128, 0, stream>>>(z_dst, g.ar, er, Nd);
    int n4 = Nd * HEADS;
    fill_k<<<(n4 + 255) / 256, 256, 0, stream>>>(m, -INFINITY, n4);
    hipMemsetAsync(den, 0, (size_t)n4 * sizeof(float), stream);
    edge_max_k<<<(E + 255) / 256, 256, 0, stream>>>(src, dst, E, el, er, m);
    edge_sum_k<<<(E + 255) / 256, 256, 0, stream>>>(src, dst, E, el, er, m, den);
    edge_agg_k<<<E, 128, 0, stream>>>(src, dst, E, el, er, m, den, z_src, acc);
}

extern "C" void kernel_launch(void* const* d_in, const int* in_sizes, int n_in,
                              void* d_out, int out_size, void* d_ws, size_t ws_size,
                              hipStream_t stream) {
    // --- unpack inputs (setup_inputs() insertion order) --------------------
    const float* x       = (const float*)d_in[0];                 // [NW,256]
    const float* doc_emb = (const float*)d_in[1];                 // [ND,256]
    Gat layer[2][4];  // [layer][ww, wwr, wd, wdr]
    int idx = 2;
    for (int L = 0; L < 2; L++)
        for (int rct = 0; rct < 4; rct++) {
            layer[L][rct].W  = (const float*)d_in[idx++];
            layer[L][rct].al = (const float*)d_in[idx++];
            layer[L][rct].ar = (const float*)d_in[idx++];
            layer[L][rct].b  = (const float*)d_in[idx++];
        }
    // lin_w (128) vs lin_b (1): disambiguate by size
    const float *lin_w, *lin_b;
    if (in_sizes[idx] == 128) { lin_w = (const float*)d_in[idx]; lin_b = (const float*)d_in[idx + 1]; }
    else                      { lin_b = (const float*)d_in[idx]; lin_w = (const float*)d_in[idx + 1]; }
    idx += 2;
    const int* ww_src  = (const int*)d_in[idx];     int E_ww  = in_sizes[idx];     idx++;
    const int* ww_dst  = (const int*)d_in[idx];                                    idx++;
    const int* wwr_src = (const int*)d_in[idx];     int E_wwr = in_sizes[idx];     idx++;
    const int* wwr_dst = (const int*)d_in[idx];                                    idx++;
    const int* wd_src  = (const int*)d_in[idx];     int E_wd  = in_sizes[idx];     idx++;
    const int* wd_dst  = (const int*)d_in[idx];                                    idx++;
    const int* wdr_src = (const int*)d_in[idx];     int E_wdr = in_sizes[idx];     idx++;
    const int* wdr_dst = (const int*)d_in[idx];                                    idx++;

    // --- workspace layout (floats) ----------------------------------------
    float* ws = (float*)d_ws;
    size_t off = 0;
    float* z_w   = ws + off; off += (size_t)NW   * ZW;
    float* z_d   = ws + off; off += (size_t)NDOC * ZW;
    float* acc_w = ws + off; off += (size_t)NW   * ZW;
    float* acc_d = ws + off; off += (size_t)NDOC * ZW;
    float* el_w  = ws + off; off += (size_t)NW   * HEADS;
    float* er_w  = ws + off; off += (size_t)NW   * HEADS;
    float* el_d  = ws + off; off += (size_t)NDOC * HEADS;
    float* er_d  = ws + off; off += (size_t)NDOC * HEADS;
    float* m_w   = ws + off; off += (size_t)NW   * HEADS;
    float* den_w = ws + off; off += (size_t)NW   * HEADS;
    float* m_d   = ws + off; off += (size_t)NDOC * HEADS;
    float* den_d = ws + off; off += (size_t)NDOC * HEADS;
    float* hw_a  = ws + off; off += (size_t)NW   * FOUT;
    float* hd_a  = ws + off; off += (size_t)NDOC * FOUT;
    float* hw_b  = ws + off; off += (size_t)NW   * FOUT;
    float* hd_b  = ws + off; off += (size_t)NDOC * FOUT;
    (void)ws_size; (void)n_in; (void)out_size;

    const float* hw = x;
    const float* hd = doc_emb;
    float* hw_next[2] = { hw_a, hw_b };
    float* hd_next[2] = { hd_a, hd_b };

    for (int L = 0; L < 2; L++) {
        int K = (L == 0) ? 256 : 128;
        const Gat& gww  = layer[L][0];
        const Gat& gwwr = layer[L][1];
        const Gat& gwd  = layer[L][2];
        const Gat& gwdr = layer[L][3];

        hipMemsetAsync(acc_w, 0, (size_t)NW   * ZW * sizeof(float), stream);
        hipMemsetAsync(acc_d, 0, (size_t)NDOC * ZW * sizeof(float), stream);

        // word -> word relations (dst = word)
        run_relation(hw, hw, NW, NW, K, gww,  ww_src,  ww_dst,  E_ww,
                     z_w, z_w, el_w, er_w, m_w, den_w, acc_w, stream);
        run_relation(hw, hw, NW, NW, K, gwwr, wwr_src, wwr_dst, E_wwr,
                     z_w, z_w, el_w, er_w, m_w, den_w, acc_w, stream);
        // doc -> word (dst = word)
        run_relation(hd, hw, NDOC, NW, K, gwdr, wdr_src, wdr_dst, E_wdr,
                     z_d, z_w, el_d, er_w, m_w, den_w, acc_w, stream);
        // word -> doc (dst = doc)
        run_relation(hw, hd, NW, NDOC, K, gwd, wd_src, wd_dst, E_wd,
                     z_w, z_d, el_w, er_d, m_d, den_d, acc_d, stream);

        finalize_k<<<((size_t)NW * FOUT + 255) / 256, 256, 0, stream>>>(
            acc_w, gww.b, gwwr.b, gwdr.b, hw_next[L], NW);
        finalize_k<<<((size_t)NDOC * FOUT + 255) / 256, 256, 0, stream>>>(
            acc_d, gwd.b, nullptr, nullptr, hd_next[L], NDOC);

        hw = hw_next[L];
        hd = hd_next[L];
    }

    final_linear_k<<<NDOC, 32, 0, stream>>>(hd, lin_w, lin_b, (float*)d_out, NDOC);
}